// GNN_EBM_34136400069077
// MI455X (gfx1250) — compile-verified
//
#include <hip/hip_runtime.h>
#include <stdint.h>

#define HID   128
#define STR   132          // activation LDS row stride (floats): conflict-free, 16B aligned
#define WSTR  260          // interleaved weight pair-row stride (floats): 2*128 + 4 pad
#define MT    64           // samples per workgroup
#define NTHR  128          // 4 waves (wave32)
#define BATCHN 8192
#define NSTEP 20

typedef float v2f __attribute__((ext_vector_type(2)));
typedef float v8f __attribute__((ext_vector_type(8)));

// LDS layout (floats)
#define OFF_IWT1 0          // 64 pair-rows * WSTR
#define OFF_IWY1 16640
#define OFF_R    33280      // 64 * STR
#define OFF_PT   41728
#define OFF_PY   50176
#define OFF_VEC  58624      // wt | wy | bT1 | bY1 | WT2 | WY2, 128 each
#define OFF_T    59392
#define OFF_Y    59456
#define OFF_GT   59520
#define OFF_GY   59584
#define LDS_FLOATS 59648    // 238592 bytes < 320KB WGP LDS

// ------------------- JAX-compatible threefry2x32 -------------------
__device__ __forceinline__ unsigned rotl32(unsigned x, int r) {
  return (x << r) | (x >> (32 - r));
}
__device__ __forceinline__ void tfround(unsigned& x0, unsigned& x1, int r) {
  x0 += x1; x1 = rotl32(x1, r); x1 ^= x0;
}
__device__ __forceinline__ void threefry2x32(unsigned k0, unsigned k1,
                                             unsigned x0, unsigned x1,
                                             unsigned& o0, unsigned& o1) {
  unsigned k2 = k0 ^ k1 ^ 0x1BD11BDAu;
  x0 += k0; x1 += k1;
  tfround(x0,x1,13); tfround(x0,x1,15); tfround(x0,x1,26); tfround(x0,x1,6);
  x0 += k1; x1 += k2 + 1u;
  tfround(x0,x1,17); tfround(x0,x1,29); tfround(x0,x1,16); tfround(x0,x1,24);
  x0 += k2; x1 += k0 + 2u;
  tfround(x0,x1,13); tfround(x0,x1,15); tfround(x0,x1,26); tfround(x0,x1,6);
  x0 += k0; x1 += k1 + 3u;
  tfround(x0,x1,17); tfround(x0,x1,29); tfround(x0,x1,16); tfround(x0,x1,24);
  x0 += k1; x1 += k2 + 4u;
  tfround(x0,x1,13); tfround(x0,x1,15); tfround(x0,x1,26); tfround(x0,x1,6);
  x0 += k2; x1 += k0 + 5u;
  o0 = x0; o1 = x1;
}

// XLA float32 erf_inv polynomial
__device__ __forceinline__ float erfinv32(float x) {
  float w = -log1pf(-x * x);
  float p;
  if (w < 5.0f) {
    w -= 2.5f;
    p = 2.81022636e-08f;
    p = fmaf(p, w, 3.43273939e-07f);
    p = fmaf(p, w, -3.5233877e-06f);
    p = fmaf(p, w, -4.39150654e-06f);
    p = fmaf(p, w, 0.00021858087f);
    p = fmaf(p, w, -0.00125372503f);
    p = fmaf(p, w, -0.00417768164f);
    p = fmaf(p, w, 0.246640727f);
    p = fmaf(p, w, 1.50140941f);
  } else {
    w = sqrtf(w) - 3.0f;
    p = -0.000200214257f;
    p = fmaf(p, w, 0.000100950558f);
    p = fmaf(p, w, 0.00134934322f);
    p = fmaf(p, w, -0.00367342844f);
    p = fmaf(p, w, 0.00573950773f);
    p = fmaf(p, w, -0.0076224613f);
    p = fmaf(p, w, 0.00943887047f);
    p = fmaf(p, w, 1.00167406f);
    p = fmaf(p, w, 2.83297682f);
  }
  return p * x;
}

__device__ __forceinline__ float bits_to_normal(unsigned bits) {
  float u = __uint_as_float((bits >> 9) | 0x3f800000u) - 1.0f;  // [0,1)
  const float lo = -0.99999994f;                                // nextafter(-1,0)
  float v = u * 2.0f + lo;
  v = fmaxf(lo, v);
  return 1.4142135f * erfinv32(v);                              // sqrt(2)*erfinv
}

// ------------------- fp32 WMMA tile GEMMs -------------------
// acc[16 x 128] += A[16 x kdim] * W[kdim x 128]; W in k-pair interleaved LDS layout.
// Forward B-frag = one contiguous b64: (W[k][n], W[k+1][n]) adjacent by construction.
__device__ __forceinline__ void mm_fwd(const float* __restrict__ aB,
                                       const float* __restrict__ wI,
                                       v8f* acc, int lane, int kdim) {
  const int mr   = (lane & 15) * STR;
  const int kh   = (lane >> 4) * 2;
  const int wof  = (lane >> 4) * WSTR + 2 * (lane & 15);
  for (int k0 = 0; k0 < kdim; k0 += 4) {
    v2f a = *(const v2f*)(aB + mr + k0 + kh);
    const float* wp = wI + (k0 >> 1) * WSTR + wof;
#pragma unroll
    for (int nt = 0; nt < 8; ++nt) {
      v2f b = *(const v2f*)(wp + nt * 32);
      acc[nt] = __builtin_amdgcn_wmma_f32_16x16x4_f32(
          false, a, false, b, (short)0, acc[nt], false, false);
    }
  }
}

// G[16 x 128] += pT * WT1^T + pY * WY1^T  (merged; weights read from interleaved layout:
// backward B-frag elements are 8B apart -> two b32 loads, immediate offsets only)
__device__ __forceinline__ void mm_bwd2(const float* __restrict__ pTB,
                                        const float* __restrict__ pYB,
                                        const float* __restrict__ w1,
                                        const float* __restrict__ w2,
                                        v8f* G, int lane) {
  const int mr = (lane & 15) * STR;
  const int kh = (lane >> 4) * 2;
  const int nc = lane & 15;
  const int jb = (nc >> 1) * WSTR + (nc & 1) + 2 * kh;  // + nt*8*WSTR + 2*k0 (immediates)
  for (int k0 = 0; k0 < HID; k0 += 4) {
    v2f aT = *(const v2f*)(pTB + mr + k0 + kh);
    v2f aY = *(const v2f*)(pYB + mr + k0 + kh);
#pragma unroll
    for (int nt = 0; nt < 8; ++nt) {
      const float* b1 = w1 + jb + nt * 8 * WSTR + 2 * k0;
      v2f bT; bT.x = b1[0]; bT.y = b1[2];
      G[nt] = __builtin_amdgcn_wmma_f32_16x16x4_f32(
          false, aT, false, bT, (short)0, G[nt], false, false);
      const float* b2 = w2 + jb + nt * 8 * WSTR + 2 * k0;
      v2f bY; bY.x = b2[0]; bY.y = b2[2];
      G[nt] = __builtin_amdgcn_wmma_f32_16x16x4_f32(
          false, aY, false, bY, (short)0, G[nt], false, false);
    }
  }
}

// row-major global -> LDS (stride STR)
__device__ __forceinline__ void stage(const float* __restrict__ g,
                                      float* __restrict__ l,
                                      int rows, int cols, int tid) {
  int total = rows * cols;
  for (int idx = tid * 4; idx < total; idx += NTHR * 4) {
    int r = idx / cols, c = idx % cols;
    float4 v = *(const float4*)(g + idx);
    *(float4*)(l + r * STR + c) = v;
  }
}

// row-major global W[k][n] (rows x 128) -> k-pair interleaved LDS
__device__ __forceinline__ void stageI(const float* __restrict__ g,
                                       float* __restrict__ l,
                                       int rows, int tid) {
  int total = rows * 128;
  for (int idx = tid * 4; idx < total; idx += NTHR * 4) {
    int k = idx >> 7, c = idx & 127;
    float4 v = *(const float4*)(g + idx);
    float* d = l + (k >> 1) * WSTR + 2 * c + (k & 1);
    d[0] = v.x; d[2] = v.y; d[4] = v.z; d[6] = v.w;
  }
}

// ------------------- setup: cT, cY and the 20 step keys -------------------
__global__ void ebm_setup(const float* __restrict__ B, float* __restrict__ ws) {
  __shared__ float sS[66];
  int tid = threadIdx.x;
  if (tid < 66) {
    float sum = 0.0f;
    for (int j = 0; j < 66; ++j) {
      if (j == tid) continue;
      if (tid == 65 && j < 64) continue;
      sum += 1.0f / (1.0f + expf(-B[tid * 66 + j]));
    }
    sS[tid] = 1.0f + sum / 66.0f;
  }
  __syncthreads();
  if (tid == 0) {
    float aT = 0.0f;
    for (int j = 0; j < 66; ++j) {
      if (j == 64) continue;
      aT += (1.0f / (1.0f + expf(-B[64 * 66 + j]))) * sS[j];
    }
    float aY = (1.0f / (1.0f + expf(-B[65 * 66 + 64]))) * sS[64];
    ws[0] = sS[64] + aT / 66.0f;   // cT
    ws[1] = sS[65] + aY / 66.0f;   // cY
    // keys = split(key(42), 20): threefry((0,42), iota(40)) -> concat(y0,y1) -> (20,2)
    unsigned y0a[20], y1a[20];
    for (int i = 0; i < 20; ++i)
      threefry2x32(0u, 42u, (unsigned)i, (unsigned)(20 + i), y0a[i], y1a[i]);
    unsigned* kk = (unsigned*)(ws + 2);
    for (int k = 0; k < 10; ++k) { kk[2 * k] = y0a[2 * k]; kk[2 * k + 1] = y0a[2 * k + 1]; }
    for (int k = 10; k < 20; ++k) {
      kk[2 * k] = y1a[2 * (k - 10)]; kk[2 * k + 1] = y1a[2 * (k - 10) + 1];
    }
  }
}

// ------------------- main persistent Langevin kernel -------------------
__global__ void __launch_bounds__(NTHR, 1)
ebm_main(const float* __restrict__ x, const float* __restrict__ t_in,
         const float* __restrict__ Win, const float* __restrict__ b_in,
         const float* __restrict__ WT1, const float* __restrict__ bT1,
         const float* __restrict__ WT2, const float* __restrict__ WY1,
         const float* __restrict__ bY1, const float* __restrict__ WY2,
         const float* __restrict__ ws, float* __restrict__ out) {
  extern __shared__ float lds[];
  const int tid  = threadIdx.x;
  const int lane = tid & 31;
  const int wave = tid >> 5;
  const int m0w  = wave * 16;
  const int row0 = blockIdx.x * MT;

  float* sIWT1 = lds + OFF_IWT1;
  float* sIWY1 = lds + OFF_IWY1;
  float* sR    = lds + OFF_R;
  float* sPT   = lds + OFF_PT;
  float* sPY   = lds + OFF_PY;
  float* sVec  = lds + OFF_VEC;
  float* sT    = lds + OFF_T;
  float* sY    = lds + OFF_Y;
  float* sGT   = lds + OFF_GT;
  float* sGY   = lds + OFF_GY;

  stageI(WT1, sIWT1, 128, tid);
  stageI(WY1, sIWY1, 128, tid);
  {
    int i = tid;                         // NTHR == 128 == HID
    sVec[i]       = Win[64 * HID + i];   // w_t
    sVec[128 + i] = Win[65 * HID + i];   // w_y
    sVec[256 + i] = bT1[i];
    sVec[384 + i] = bY1[i];
    sVec[512 + i] = WT2[i];
    sVec[640 + i] = WY2[i];
  }
  stage(x + (size_t)row0 * 64, sPT, 64, 64, tid);  // x tile (A, K=64)
  stageI(Win, sPY, 64, tid);                       // W_in[0:64] interleaved (B)
  if (tid < MT) { sT[tid] = t_in[row0 + tid]; sY[tid] = 0.0f; }
  __syncthreads();

  const float cTs = ws[0], cYs = ws[1];
  const unsigned* keys = (const unsigned*)(ws + 2);

  const int ncol = lane & 15, hi = lane >> 4;
  float rWt[8], rWy[8], rBT1[8], rBY1[8], rWT2[8], rWY2[8];
#pragma unroll
  for (int nt = 0; nt < 8; ++nt) {
    int n = nt * 16 + ncol;
    rWt[nt]  = sVec[n];        rWy[nt]  = sVec[128 + n];
    rBT1[nt] = sVec[256 + n];  rBY1[nt] = sVec[384 + n];
    rWT2[nt] = sVec[512 + n];  rWY2[nt] = sVec[640 + n];
  }

  // cacc = x @ W_in[0:64] + b_in, kept in registers (C/D layout) for the whole kernel
  v8f cacc[8];
#pragma unroll
  for (int nt = 0; nt < 8; ++nt) cacc[nt] = (v8f){0,0,0,0,0,0,0,0};
  mm_fwd(sPT + m0w * STR, sPY, cacc, lane, 64);
#pragma unroll
  for (int nt = 0; nt < 8; ++nt) {
    float bn = b_in[nt * 16 + ncol];
#pragma unroll
    for (int v = 0; v < 8; ++v) cacc[nt][v] += bn;
  }
  __syncthreads();  // sPT/sPY reuse below

#pragma unroll 1
  for (int step = 0; step < NSTEP; ++step) {
    // ---- r = relu(c + t*w_t + y*w_y): wave-private rows, no barrier needed
    float tvv[8], yvv[8];
#pragma unroll
    for (int v = 0; v < 8; ++v) {
      int m = m0w + v + 8 * hi;
      tvv[v] = sT[m]; yvv[v] = sY[m];
#pragma unroll
      for (int nt = 0; nt < 8; ++nt) {
        float h = cacc[nt][v] + tvv[v] * rWt[nt] + yvv[v] * rWy[nt];
        sR[m * STR + nt * 16 + ncol] = fmaxf(h, 0.0f);
      }
    }

    // ---- forward T: aT = cT*(r@WT1)+bT1 ; pT = (aT>0)? cT*WT2 : 0
    {
      v8f a[8];
#pragma unroll
      for (int nt = 0; nt < 8; ++nt) a[nt] = (v8f){0,0,0,0,0,0,0,0};
      mm_fwd(sR + m0w * STR, sIWT1, a, lane, HID);
#pragma unroll
      for (int nt = 0; nt < 8; ++nt)
#pragma unroll
        for (int v = 0; v < 8; ++v) {
          float aT = cTs * a[nt][v] + rBT1[nt];
          sPT[(m0w + v + 8 * hi) * STR + nt * 16 + ncol] =
              (aT > 0.0f) ? cTs * rWT2[nt] : 0.0f;
        }
    }
    // ---- forward Y
    {
      v8f a[8];
#pragma unroll
      for (int nt = 0; nt < 8; ++nt) a[nt] = (v8f){0,0,0,0,0,0,0,0};
      mm_fwd(sR + m0w * STR, sIWY1, a, lane, HID);
#pragma unroll
      for (int nt = 0; nt < 8; ++nt)
#pragma unroll
        for (int v = 0; v < 8; ++v) {
          float aY = cYs * a[nt][v] + rBY1[nt];
          sPY[(m0w + v + 8 * hi) * STR + nt * 16 + ncol] =
              (aY > 0.0f) ? cYs * rWY2[nt] : 0.0f;
        }
    }

    // ---- backward (merged): G = pT@WT1^T + pY@WY1^T  (wave-local rows)
    v8f G[8];
#pragma unroll
    for (int nt = 0; nt < 8; ++nt) G[nt] = (v8f){0,0,0,0,0,0,0,0};
    mm_bwd2(sPT + m0w * STR, sPY + m0w * STR, sIWT1, sIWY1, G, lane);

    // ---- mask by (h0>0) (recomputed from registers) and reduce vs w_t / w_y
    float pt[8], py[8];
#pragma unroll
    for (int v = 0; v < 8; ++v) { pt[v] = 0.0f; py[v] = 0.0f; }
#pragma unroll
    for (int nt = 0; nt < 8; ++nt)
#pragma unroll
      for (int v = 0; v < 8; ++v) {
        float h = cacc[nt][v] + tvv[v] * rWt[nt] + yvv[v] * rWy[nt];
        float g = (h > 0.0f) ? G[nt][v] : 0.0f;
        pt[v] += g * rWt[nt];
        py[v] += g * rWy[nt];
      }
#pragma unroll
    for (int v = 0; v < 8; ++v)
#pragma unroll
      for (int msk = 1; msk <= 8; msk <<= 1) {
        pt[v] += __shfl_xor(pt[v], msk, 32);
        py[v] += __shfl_xor(py[v], msk, 32);
      }
    if (ncol == 0) {
#pragma unroll
      for (int v = 0; v < 8; ++v) {
        sGT[m0w + 8 * hi + v] = pt[v];
        sGY[m0w + 8 * hi + v] = py[v];
      }
    }
    __syncthreads();

    // ---- Langevin update with exact JAX threefry noise; thread (m,j)
    {
      int m = tid >> 1, j = tid & 1;
      unsigned i = 2u * (unsigned)(row0 + m) + (unsigned)j;  // flat idx into (8192,2)
      unsigned c0 = (i < 8192u) ? i : i - 8192u;
      unsigned o0, o1;
      threefry2x32(keys[2 * step], keys[2 * step + 1], c0, c0 + 8192u, o0, o1);
      unsigned bits = (i < 8192u) ? o0 : o1;
      float nrm = bits_to_normal(bits);
      float g  = j ? sGY[m] : sGT[m];
      float zv = j ? sY[m]  : sT[m];
      zv = (zv - 0.005f * g) + sqrtf(0.01f) * nrm;
      if (j) sY[m] = zv; else sT[m] = zv;
    }
    __syncthreads();
  }

  if (tid < MT) out[row0 + tid] = sY[tid];
}

extern "C" void kernel_launch(void* const* d_in, const int* in_sizes, int n_in,
                              void* d_out, int out_size, void* d_ws, size_t ws_size,
                              hipStream_t stream) {
  (void)in_sizes; (void)n_in; (void)out_size; (void)ws_size;
  const float* x    = (const float*)d_in[0];
  const float* t_in = (const float*)d_in[1];
  const float* Bp   = (const float*)d_in[2];
  const float* Win  = (const float*)d_in[3];
  const float* b_in = (const float*)d_in[4];
  const float* WT1  = (const float*)d_in[5];
  const float* bT1  = (const float*)d_in[6];
  const float* WT2  = (const float*)d_in[7];
  // d_in[8] = bT2 (vanishes in gradient)
  const float* WY1  = (const float*)d_in[9];
  const float* bY1  = (const float*)d_in[10];
  const float* WY2  = (const float*)d_in[11];
  // d_in[12] = bY2 (vanishes in gradient)
  float* ws  = (float*)d_ws;
  float* out = (float*)d_out;

  ebm_setup<<<dim3(1), dim3(128), 0, stream>>>(Bp, ws);
  ebm_main<<<dim3(BATCHN / MT), dim3(NTHR), LDS_FLOATS * sizeof(float), stream>>>(
      x, t_in, Win, b_in, WT1, bT1, WT2, WY1, bY1, WY2, ws, out);
}